// ChamferDistance_1726576856987
// MI455X (gfx1250) — compile-verified
//
#include <hip/hip_runtime.h>

typedef float v2f __attribute__((ext_vector_type(2)));
typedef float v8f __attribute__((ext_vector_type(8)));

#define TILES 4
#define WAVES 8
#define BLOCK (WAVES * 32)
#define ROWS_PER_WAVE (TILES * 16)        // 64
#define ROWS_PER_WG   (WAVES * ROWS_PER_WAVE) // 512
#define NBATCH 8
#define NPTS 8192

// Single v_min_num_f32 without the llvm.minnum canonicalization (v_max x,x)
// the compiler otherwise wraps around non-provably-canonical WMMA results.
__device__ __forceinline__ float fmin_raw(float a, float b) {
    float r;
    asm("v_min_num_f32 %0, %1, %2" : "=v"(r) : "v"(a), "v"(b));
    return r;
}

// Computes out[b][n] = min_m ||A[b][n] - B[b][m]||^2 for n in this block's rows.
// gridDim = (NPTS/ROWS_PER_WG, NBATCH, 2); z selects role (dist1 vs dist2).
__global__ __launch_bounds__(BLOCK)
void chamfer_wmma_kernel(const float* __restrict__ xyz1,
                         const float* __restrict__ xyz2,
                         float* __restrict__ out) {
    const int role = blockIdx.z;
    const float* __restrict__ Apts = role ? xyz2 : xyz1;
    const float* __restrict__ Bpts = role ? xyz1 : xyz2;
    float* __restrict__ o = out + (size_t)role * (NBATCH * (size_t)NPTS);

    const int b    = blockIdx.y;
    const int lane = threadIdx.x & 31;
    const int wave = threadIdx.x >> 5;
    const int half = lane >> 4;       // 0: lanes 0-15, 1: lanes 16-31
    const int l16  = lane & 15;
    const size_t base = (size_t)b * NPTS * 3;
    const int rbase = blockIdx.x * ROWS_PER_WG + wave * ROWS_PER_WAVE;

    // ---- Load A tiles once: A row n = (-2x, -2y, -2z, 1) ----
    // A 16x4 f32 layout: lanes 0-15 hold (K0,K1), lanes 16-31 hold (K2,K3), row = lane%16.
    v2f   a[TILES];
    float sqa[TILES];
#pragma unroll
    for (int t = 0; t < TILES; ++t) {
        const int row = rbase + t * 16 + l16;
        const float* p = Apts + base + (size_t)row * 3;
        const float x = p[0], y = p[1], z = p[2];
        sqa[t] = x * x + y * y + z * z;      // added after the min-reduction
        a[t][0] = half ? (-2.0f * z) : (-2.0f * x);
        a[t][1] = half ? 1.0f        : (-2.0f * y);
    }

    // Running minima of (||x2||^2 - 2<x1,x2>) per (tile, output-VGPR)
    v8f runm[TILES];
#pragma unroll
    for (int t = 0; t < TILES; ++t)
#pragma unroll
        for (int v = 0; v < 8; ++v) runm[t][v] = 3.402823466e38f;

    // ---- Sweep all M columns in 16-wide chunks ----
#pragma unroll 2
    for (int m0 = 0; m0 < NPTS; m0 += 16) {
        const float* q = Bpts + base + (size_t)(m0 + l16) * 3;
        const float x = q[0], y = q[1], z = q[2];
        __builtin_prefetch(q + 4 * 16 * 3, 0, 1);   // 4 chunks ahead -> global_prefetch_b8
        const float sqb = x * x + y * y + z * z;
        // B 4x16 f32 layout: lanes 0-15 hold (K0,K1), lanes 16-31 hold (K2,K3), col = lane%16.
        v2f bm;
        bm[0] = half ? z   : x;
        bm[1] = half ? sqb : y;

#pragma unroll
        for (int t = 0; t < TILES; ++t) {
            v8f c = {};   // fresh zero accumulator (SRC2 inline 0)
            v8f d = __builtin_amdgcn_wmma_f32_16x16x4_f32(
                false, a[t], false, bm, (short)0, c, false, false);
#pragma unroll
            for (int v = 0; v < 8; ++v)
                runm[t][v] = fmin_raw(runm[t][v], d[v]);
        }
    }

    // ---- Reduce across the 16 lanes of each half (one row per half per VGPR) ----
#pragma unroll
    for (int t = 0; t < TILES; ++t) {
#pragma unroll
        for (int v = 0; v < 8; ++v) {
            float r = runm[t][v];
            r = fmin_raw(r, __shfl_xor(r, 1, 32));
            r = fmin_raw(r, __shfl_xor(r, 2, 32));
            r = fmin_raw(r, __shfl_xor(r, 4, 32));
            r = fmin_raw(r, __shfl_xor(r, 8, 32));
            const float slo = __shfl(sqa[t], v,     32); // ||A[row_lo]||^2
            const float shi = __shfl(sqa[t], v + 8, 32); // ||A[row_hi]||^2
            const int row_lo = rbase + t * 16 + v;
            if (lane == 0)  o[(size_t)b * NPTS + row_lo]     = r + slo;
            if (lane == 16) o[(size_t)b * NPTS + row_lo + 8] = r + shi;
        }
    }
}

extern "C" void kernel_launch(void* const* d_in, const int* in_sizes, int n_in,
                              void* d_out, int out_size, void* d_ws, size_t ws_size,
                              hipStream_t stream) {
    (void)in_sizes; (void)n_in; (void)out_size; (void)d_ws; (void)ws_size;
    const float* xyz1 = (const float*)d_in[0];
    const float* xyz2 = (const float*)d_in[1];
    float* out = (float*)d_out;

    dim3 grid(NPTS / ROWS_PER_WG, NBATCH, 2);
    dim3 block(BLOCK, 1, 1);
    chamfer_wmma_kernel<<<grid, block, 0, stream>>>(xyz1, xyz2, out);
}